// Caser_64407329571691
// MI455X (gfx1250) — compile-verified
//
#include <hip/hip_runtime.h>
#include <hip/hip_bf16.h>

typedef __attribute__((ext_vector_type(2))) float v2f;
typedef __attribute__((ext_vector_type(8))) float v8f;

#define CB   2048
#define CL   5
#define CD   64
#define CNH  16
#define CNV  4
#define CIL  1000
#define VH_DIM (CNV * CD + CNH)   // 272
#define ZU_DIM (2 * CD)           // 128

// ---------------------------------------------------------------------------
// Kernel 1: per-batch-row prep (embedding gather, vertical conv, horizontal
// conv + masked max + relu, fc1, concat with user embedding) -> zu[b][128].
// Tiny fraction of total FLOPs/bytes; plain VALU, one 64-thread block per b.
// ---------------------------------------------------------------------------
__global__ __launch_bounds__(64)
void caser_prep(const int* __restrict__ seq, const int* __restrict__ user,
                const float* __restrict__ item_emb, const float* __restrict__ user_emb,
                const float* __restrict__ vw, const float* __restrict__ vb,
                const float* __restrict__ hw, const float* __restrict__ hb,
                const int* __restrict__ heights,
                const float* __restrict__ fc1_w, const float* __restrict__ fc1_b,
                float* __restrict__ zu) {
  __shared__ float s_emb[CL][CD];
  __shared__ float s_vh[VH_DIM];

  const int b = blockIdx.x;
  const int t = threadIdx.x;          // 0..63 (= d index / output index)

  // Gather item embeddings for the sequence: emb[l][d]
  for (int l = 0; l < CL; ++l)
    s_emb[l][t] = item_emb[(size_t)seq[b * CL + l] * CD + t];
  __syncthreads();

  // Vertical convolution: ver[f*D+d] = sum_l emb[l][d]*vw[f][l] + vb[f]
  for (int f = 0; f < CNV; ++f) {
    float s = vb[f];
    for (int l = 0; l < CL; ++l)
      s += s_emb[l][t] * vw[f * CL + l];
    s_vh[f * CD + t] = s;
  }

  // Horizontal convolution (threads 0..15, one filter each):
  // scores[f][tt] = sum_{i: tt+i<L} sum_d emb[tt+i][d]*hw[f][i][d] + hb[f]
  // valid tt <= L - heights[f]; hor[f] = relu(max over valid tt)
  if (t < CNH) {
    const int h = heights[t];
    const int tmax = CL - h;          // tt = 0..tmax, tt=0 always valid
    float best = -3.402823466e+38f;
    for (int tt = 0; tt <= tmax; ++tt) {
      float s = hb[t];
      for (int i = 0; i + tt < CL; ++i) {
        const float* hwrow = hw + ((size_t)t * CL + i) * CD;
        for (int d = 0; d < CD; ++d)
          s += s_emb[tt + i][d] * hwrow[d];
      }
      best = fmaxf(best, s);
    }
    s_vh[CNV * CD + t] = fmaxf(best, 0.0f);
  }
  __syncthreads();

  // fc1: z[j] = relu(sum_k vh[k]*fc1_w[j][k] + fc1_b[j]); zu = [z, u]
  float z = fc1_b[t];
  const float* wrow = fc1_w + (size_t)t * VH_DIM;
  for (int k = 0; k < VH_DIM; ++k)
    z += s_vh[k] * wrow[k];
  zu[(size_t)b * ZU_DIM + t]      = fmaxf(z, 0.0f);
  zu[(size_t)b * ZU_DIM + CD + t] = user_emb[(size_t)user[b] * CD + t];
}

// ---------------------------------------------------------------------------
// Kernel 2: the memory-bound gather-matvec, on the matrix pipe.
// res[b,i] = dot(W2[items[b,i]], zu[b]) + b2[items[b,i]]
// One wave per 16-item tile. A = 16 gathered W2 rows, B = zu broadcast into
// all 16 columns, accumulated with V_WMMA_F32_16X16X4_F32 (f32, K stepped 4).
// 8 waves (256 threads) per block, 128 items per block; W2 (51 MB) is
// L2-resident on the 192 MB L2, so this should run at L2 bandwidth.
// ---------------------------------------------------------------------------
__global__ __launch_bounds__(256)
void caser_score(const int* __restrict__ items, const float* __restrict__ W2,
                 const float* __restrict__ b2, const float* __restrict__ zu,
                 float* __restrict__ out) {
  __shared__ float s_zu[ZU_DIM];

  const int b   = blockIdx.x;
  const int tid = threadIdx.x;        // 0..255
  if (tid < ZU_DIM)
    s_zu[tid] = zu[(size_t)b * ZU_DIM + tid];
  __syncthreads();

  const int wave = tid >> 5;          // 0..7
  const int lane = tid & 31;
  const int i0   = blockIdx.y * 128 + wave * 16;   // first item of this tile
  if (i0 >= CIL) return;              // wave-uniform exit: EXEC stays all-1s

  const int m     = lane & 15;        // A-matrix row within tile
  const int khalf = lane >> 4;        // 0: K=0,1  1: K=2,3 per WMMA step
  const int item_i = (i0 + m < CIL) ? (i0 + m) : (CIL - 1);   // clamp tail
  const int idx    = items[(size_t)b * CIL + item_i];
  const float* __restrict__ row = W2 + (size_t)idx * ZU_DIM;

  v8f acc = {};
  // K = 128 in steps of 4 -> 32 chained f32 WMMAs, full f32 precision (RNE).
  for (int k0 = 0; k0 < ZU_DIM; k0 += 4) {
    v2f a;
    a.x = row[k0 + 2 * khalf];
    a.y = row[k0 + 2 * khalf + 1];
    v2f bb;                           // broadcast zu into every B column
    bb.x = s_zu[k0 + 2 * khalf];
    bb.y = s_zu[k0 + 2 * khalf + 1];
    acc = __builtin_amdgcn_wmma_f32_16x16x4_f32(
        /*neg_a=*/false, a, /*neg_b=*/false, bb,
        /*c_mod=*/(short)0, acc, /*reuse_a=*/false, /*reuse_b=*/false);
  }

  // With broadcast-B every D column equals the dot product:
  //   lane 0  holds rows 0..7  in acc[0..7], lane 16 holds rows 8..15.
  if (m == 0) {
    const int base = i0 + khalf * 8;
    #pragma unroll
    for (int r = 0; r < 8; ++r) {
      const int i = base + r;
      if (i < CIL) {
        const int id2 = items[(size_t)b * CIL + i];
        out[(size_t)b * CIL + i] = acc[r] + b2[id2];
      }
    }
  }
}

extern "C" void kernel_launch(void* const* d_in, const int* in_sizes, int n_in,
                              void* d_out, int out_size, void* d_ws, size_t ws_size,
                              hipStream_t stream) {
  const int*   seq      = (const int*)  d_in[0];
  const int*   user     = (const int*)  d_in[1];
  const int*   items    = (const int*)  d_in[2];
  const float* item_emb = (const float*)d_in[3];
  const float* user_emb = (const float*)d_in[4];
  const float* vw       = (const float*)d_in[5];
  const float* vb       = (const float*)d_in[6];
  const float* hw       = (const float*)d_in[7];
  const float* hb       = (const float*)d_in[8];
  const int*   heights  = (const int*)  d_in[9];
  const float* fc1_w    = (const float*)d_in[10];
  const float* fc1_b    = (const float*)d_in[11];
  const float* W2       = (const float*)d_in[12];
  const float* b2       = (const float*)d_in[13];
  float*       out      = (float*)d_out;
  float*       zu       = (float*)d_ws;            // B*128 f32 = 1 MB scratch

  caser_prep<<<dim3(CB), dim3(64), 0, stream>>>(
      seq, user, item_emb, user_emb, vw, vb, hw, hb, heights, fc1_w, fc1_b, zu);

  // grid.y * 128 items per block covers IL=1000 (8*128 = 1024, tail masked)
  caser_score<<<dim3(CB, 8), dim3(256), 0, stream>>>(items, W2, b2, zu, out);
}